// LoRAConv3d_27118423507086
// MI455X (gfx1250) — compile-verified
//
#include <hip/hip_runtime.h>

typedef __attribute__((ext_vector_type(2))) float v2f;
typedef __attribute__((ext_vector_type(8))) float v8f;

#define BATCH  2
#define CINC   64
#define COUTC  128
#define RANKC  16
#define NTOT   144            // COUT + RANK
#define NTILES 9              // NTOT / 16
#define DIMS   32             // D = H = W
#define KTAPS  27             // 3*3*3
#define KPC    28             // per-channel K padded to multiple of 4
#define KSTEPS 7              // KPC / 4
#define KTOTSTEPS (CINC * KSTEPS)   // 448 WMMA k-steps
#define CB     8              // channels staged per LDS pass
#define XROW   34             // 32 + 2 halo
#define NTHR1  288            // 9 waves

// ---------------------------------------------------------------------------
// Kernel 0: pack weight[128,64,3,3,3] + lora_A[16,1728] directly into per-lane
// WMMA B-fragment order:  WtP[ntile][kstep][lane] = float2{B[k0][n], B[k0+1][n]}
// with n = ntile*16 + lane%16, k0 = (kstep%7)*4 + 2*(lane/16) within channel
// c = kstep/7 (tap 27 is a zero pad). Inner loop B load = one coalesced b64.
// ---------------------------------------------------------------------------
__device__ __forceinline__ float wfetch(const float* __restrict__ weight,
                                        const float* __restrict__ lora_A,
                                        int n, int c, int off) {
    if (off >= KTAPS) return 0.0f;
    int src = c * KTAPS + off;                       // both: c outer, taps inner
    return (n < COUTC) ? weight[n * (CINC * KTAPS) + src]
                       : lora_A[(n - COUTC) * (CINC * KTAPS) + src];
}

__global__ void pack_weights(const float* __restrict__ weight,
                             const float* __restrict__ lora_A,
                             v2f* __restrict__ WtP) {
    int f = blockIdx.x * blockDim.x + threadIdx.x;   // float2 element index
    if (f >= NTILES * KTOTSTEPS * 32) return;
    int lane  = f & 31;
    int t     = f >> 5;
    int kstep = t % KTOTSTEPS;
    int nt    = t / KTOTSTEPS;
    int c     = kstep / KSTEPS;
    int ks    = kstep - c * KSTEPS;
    int n     = nt * 16 + (lane & 15);
    int off0  = ks * 4 + ((lane >> 4) << 1);
    v2f v;
    v.x = wfetch(weight, lora_A, n, c, off0);
    v.y = wfetch(weight, lora_A, n, c, off0 + 1);
    WtP[f] = v;
}

// ---------------------------------------------------------------------------
// Kernel 1: implicit-GEMM conv. One block per (b,z,y): M = 32 (W row, two
// 16-wide subtiles), N = 144 (9 waves x 16), K = 1792 padded.
// A fragments staged pre-unfolded in LDS (one ds_load_b64 each, bank-clean);
// B fragments are a coalesced sequential global_load_b64 stream.
// ---------------------------------------------------------------------------
__launch_bounds__(NTHR1)
__global__ void conv_gemm(const float* __restrict__ x,
                          const v2f* __restrict__ WtP,
                          const float* __restrict__ bias,
                          float* __restrict__ out,
                          float* __restrict__ low) {
    __shared__ float xs[CB * 9 * XROW];    // compact halo rows [c][i*3+j][xi]
    __shared__ float af[CB * 32 * KPC];    // unfolded A [c][m][k], k contiguous

    const int tid  = threadIdx.x;
    const int bz   = blockIdx.x;           // 0..2047
    const int y    = bz & 31;
    const int z    = (bz >> 5) & 31;
    const int b    = bz >> 10;

    const int wave = tid >> 5;             // 0..8 -> N tile
    const int lane = tid & 31;
    const int nl   = lane & 15;
    const int hi   = lane >> 4;

    v8f acc0 = {0.f,0.f,0.f,0.f,0.f,0.f,0.f,0.f};   // m0 = 0
    v8f acc1 = {0.f,0.f,0.f,0.f,0.f,0.f,0.f,0.f};   // m0 = 16

    const v2f* af_v2 = (const v2f*)af;

    for (int cb = 0; cb < CINC; cb += CB) {
        __syncthreads();                   // previous chunk's af fully consumed
        // --- stage compact halo rows: CB ch x 9 (i,j) x 34 (x=-1..32) ------
        for (int e = tid; e < CB * 9 * XROW; e += NTHR1) {
            int c  = e / (9 * XROW);
            int r  = e - c * (9 * XROW);
            int ij = r / XROW;
            int xi = r - ij * XROW;
            int i  = ij / 3, j = ij - i * 3;
            int zz = z + i - 1, yy = y + j - 1, xx = xi - 1;
            float v = 0.0f;
            if ((unsigned)zz < DIMS && (unsigned)yy < DIMS && (unsigned)xx < DIMS)
                v = x[(((b * CINC + cb + c) * DIMS + zz) * DIMS + yy) * DIMS + xx];
            xs[e] = v;
        }
        __syncthreads();
        // --- expand LDS->LDS into fragment order af[c][m][k] ---------------
        for (int e = tid; e < CB * 32 * KPC; e += NTHR1) {
            int c = e / (32 * KPC);
            int r = e - c * (32 * KPC);
            int m = r / KPC;
            int k = r - m * KPC;
            float v = 0.0f;
            if (k < KTAPS) {
                int ij = k / 3, l = k - ij * 3;      // ij = i*3+j
                v = xs[c * (9 * XROW) + ij * XROW + m + l];
            }
            af[e] = v;
        }
        __syncthreads();
        // --- GEMM over this channel chunk ----------------------------------
        #pragma unroll 1
        for (int cl = 0; cl < CB; ++cl) {
            const int c = cb + cl;
            // A rows for this lane: v2f index (m row)*14 + 2*ks + hi
            const v2f* arow0 = af_v2 + (cl * 32 + nl) * (KPC / 2) + hi;
            const v2f* arow1 = arow0 + 16 * (KPC / 2);
            // B stream: WtP[(wave*448 + c*7 + ks)*32 + lane]
            const v2f* bp = WtP + ((size_t)(wave * KTOTSTEPS + c * KSTEPS)) * 32 + lane;
            __builtin_prefetch((const void*)(bp + KSTEPS * 32), 0, 1);  // next ch
            #pragma unroll
            for (int ks = 0; ks < KSTEPS; ++ks) {
                v2f a0 = arow0[2 * ks];            // ds_load_b64, no remix
                v2f a1 = arow1[2 * ks];
                v2f bf = bp[ks * 32];              // coalesced global_load_b64
                acc0 = __builtin_amdgcn_wmma_f32_16x16x4_f32(
                           false, a0, false, bf, (short)0, acc0, false, false);
                acc1 = __builtin_amdgcn_wmma_f32_16x16x4_f32(
                           false, a1, false, bf, (short)0, acc1, false, false);
            }
        }
    }

    // Epilogue. D layout: VGPR v -> M=v (lanes 0-15) / v+8 (lanes 16-31), N=lane%16.
    // Each lane's 8 elements are M-consecutive -> two b128 stores per tile.
    if (wave < 8) {
        const int n   = wave * 16 + nl;
        const float bs = bias[n];
        float* op = &out[((((b * COUTC + n) * DIMS + z) * DIMS + y) * DIMS) + 8 * hi];
        float4 s0 = make_float4(acc0[0] + bs, acc0[1] + bs, acc0[2] + bs, acc0[3] + bs);
        float4 s1 = make_float4(acc0[4] + bs, acc0[5] + bs, acc0[6] + bs, acc0[7] + bs);
        float4 s2 = make_float4(acc1[0] + bs, acc1[1] + bs, acc1[2] + bs, acc1[3] + bs);
        float4 s3 = make_float4(acc1[4] + bs, acc1[5] + bs, acc1[6] + bs, acc1[7] + bs);
        *(float4*)(op)      = s0;
        *(float4*)(op + 4)  = s1;
        *(float4*)(op + 16) = s2;
        *(float4*)(op + 20) = s3;
    } else {
        const int r = nl;                    // low-rank channel
        float* lp = &low[((((b * RANKC + r) * DIMS + z) * DIMS + y) * DIMS) + 8 * hi];
        float4 s0 = make_float4(acc0[0], acc0[1], acc0[2], acc0[3]);
        float4 s1 = make_float4(acc0[4], acc0[5], acc0[6], acc0[7]);
        float4 s2 = make_float4(acc1[0], acc1[1], acc1[2], acc1[3]);
        float4 s3 = make_float4(acc1[4], acc1[5], acc1[6], acc1[7]);
        *(float4*)(lp)      = s0;
        *(float4*)(lp + 4)  = s1;
        *(float4*)(lp + 16) = s2;
        *(float4*)(lp + 20) = s3;
    }
}

// ---------------------------------------------------------------------------
// Kernel 2: out += 2.0 * lora_B @ boxsum3x3x3(low).  One block per (b,z,y).
// ---------------------------------------------------------------------------
__launch_bounds__(256)
__global__ void lora_fold_add(const float* __restrict__ low,
                              const float* __restrict__ lora_B,
                              float* __restrict__ out) {
    __shared__ float lb[RANKC * DIMS];     // box-summed low for this (b,z,y)

    const int tid = threadIdx.x;
    const int bz  = blockIdx.x;
    const int y   = bz & 31;
    const int z   = (bz >> 5) & 31;
    const int b   = bz >> 10;

    for (int e = tid; e < RANKC * DIMS; e += 256) {
        int r  = e >> 5;
        int xx = e & 31;
        float s = 0.0f;
        const float* lr = &low[(size_t)(b * RANKC + r) * DIMS * DIMS * DIMS];
        #pragma unroll
        for (int i = -1; i <= 1; ++i) {
            int zz = z + i;
            if ((unsigned)zz >= DIMS) continue;
            #pragma unroll
            for (int j = -1; j <= 1; ++j) {
                int yy = y + j;
                if ((unsigned)yy >= DIMS) continue;
                const float* row = &lr[(zz * DIMS + yy) * DIMS];
                #pragma unroll
                for (int l = -1; l <= 1; ++l) {
                    int xw = xx + l;
                    if ((unsigned)xw < DIMS) s += row[xw];
                }
            }
        }
        lb[e] = s;
    }
    __syncthreads();

    const int xx = tid & 31;
    const int og = tid >> 5;               // 0..7, each handles 16 out channels
    #pragma unroll
    for (int oo = 0; oo < 16; ++oo) {
        int o = og * 16 + oo;
        float s = 0.0f;
        #pragma unroll
        for (int r = 0; r < RANKC; ++r)
            s += lora_B[o * RANKC + r] * lb[r * DIMS + xx];
        out[(((b * COUTC + o) * DIMS + z) * DIMS + y) * DIMS + xx] += 2.0f * s;
    }
}

// ---------------------------------------------------------------------------
extern "C" void kernel_launch(void* const* d_in, const int* in_sizes, int n_in,
                              void* d_out, int out_size, void* d_ws, size_t ws_size,
                              hipStream_t stream) {
    const float* x      = (const float*)d_in[0];
    const float* weight = (const float*)d_in[1];
    const float* bias   = (const float*)d_in[2];
    const float* lora_A = (const float*)d_in[3];
    const float* lora_B = (const float*)d_in[4];
    float* out = (float*)d_out;

    v2f*   WtP = (v2f*)d_ws;                                  // 129024 float2 (~1.0 MB)
    float* low = (float*)d_ws + (size_t)NTILES * KTOTSTEPS * 32 * 2;  // ~4.2 MB

    const int packN = NTILES * KTOTSTEPS * 32;                // 129024 float2 elems
    pack_weights<<<(packN + 255) / 256, 256, 0, stream>>>(weight, lora_A, WtP);

    const int blocks = BATCH * DIMS * DIMS;                   // 2048: one per (b,z,y)
    conv_gemm<<<blocks, NTHR1, 0, stream>>>(x, WtP, bias, out, low);
    lora_fold_add<<<blocks, 256, 0, stream>>>(low, lora_B, out);
}